// GraphTransformerGraphDistribution_75196287418544
// MI455X (gfx1250) — compile-verified
//
#include <hip/hip_runtime.h>
#include <math.h>

// B=256, N=128, NF=EF=5, ED=129
typedef __attribute__((ext_vector_type(2))) float v2f;
typedef __attribute__((ext_vector_type(8))) float v8f;

__device__ __forceinline__ float sigmoidf(float x) { return 1.0f / (1.0f + __expf(-x)); }

// ---------------------------------------------------------------------------
// Kernel 1: nodes_emb / edges_emb = sigmoid(embedding @ W + b)   [256 x 5] each
// ---------------------------------------------------------------------------
__global__ __launch_bounds__(32) void emb_kernel(
    const float* __restrict__ emb,
    const float* __restrict__ Wn, const float* __restrict__ bn,
    const float* __restrict__ We, const float* __restrict__ be,
    float* __restrict__ wsNE, float* __restrict__ wsEE) {
  const int b = blockIdx.x, t = threadIdx.x;
  if (t < 10) {
    const int col = t % 5;
    const float* W = (t < 5) ? Wn : We;
    float acc = (t < 5) ? bn[col] : be[col];
    const float* e = emb + (size_t)b * 129;
    for (int k = 0; k < 129; ++k) acc += e[k] * W[k * 5 + col];
    const float r = sigmoidf(acc);
    if (t < 5) wsNE[b * 5 + col] = r; else wsEE[b * 5 + col] = r;
  }
}

// ---------------------------------------------------------------------------
// Kernel 2: Laplacian from adjacency (argmax(edges) > 0).  Block = (i, b).
// ---------------------------------------------------------------------------
__global__ __launch_bounds__(128) void lap_kernel(
    const float* __restrict__ edges, float* __restrict__ Ag) {
  __shared__ float red[128];
  const int i = blockIdx.x, b = blockIdx.y, j = threadIdx.x;
  const float* e = edges + ((((size_t)b * 128 + i) * 128) + j) * 5;
  float best = e[0]; int arg = 0;
  for (int k = 1; k < 5; ++k) { float v = e[k]; if (v > best) { best = v; arg = k; } }
  const float a = (arg > 0) ? 1.0f : 0.0f;
  red[j] = a;
  __syncthreads();
  for (int off = 64; off > 0; off >>= 1) {
    if (j < off) red[j] += red[j + off];
    __syncthreads();
  }
  const float deg = red[0];
  Ag[(((size_t)b * 128 + i) * 128) + j] = (j == i) ? (deg - a) : -a;
}

// ---------------------------------------------------------------------------
// Kernel 3: batched symmetric eigh via parallel-order cyclic Jacobi.
// One block per matrix; A,V resident in LDS (stride 129 -> bank-conflict-free
// column updates).  12 sweeps x 127 tournament rounds, 64 disjoint pairs/round.
// Eigenvalues sorted ascending (index tie-break) to match jnp.linalg.eigh.
// ---------------------------------------------------------------------------
__global__ __launch_bounds__(128) void jacobi_kernel(
    const float* __restrict__ Ag, float* __restrict__ Vg, float* __restrict__ evg) {
  __shared__ float As[128 * 129];
  __shared__ float Vs[128 * 129];
  __shared__ int   pA[64], qA[64];
  __shared__ float cA[64], sA[64];
  __shared__ float ev[128];
  const int b = blockIdx.x;
  const int t = threadIdx.x;

  for (int row = 0; row < 128; ++row) {
    As[row * 129 + t] = Ag[((size_t)b * 128 + row) * 128 + t];
    Vs[row * 129 + t] = (row == t) ? 1.0f : 0.0f;
  }
  __syncthreads();

  for (int sweep = 0; sweep < 12; ++sweep) {
    for (int r = 0; r < 127; ++r) {
      // Phase A: 64 disjoint rotations (round-robin tournament schedule)
      if (t < 64) {
        int p, q;
        if (t == 0) { p = 127; q = r % 127; }
        else        { p = (r + t) % 127; q = (r + 127 - t) % 127; }
        if (p > q) { int tmp = p; p = q; q = tmp; }
        const float apq = As[p * 129 + q];
        float c = 1.0f, s = 0.0f;
        if (fabsf(apq) > 1e-12f) {
          const float tau = (As[q * 129 + q] - As[p * 129 + p]) / (2.0f * apq);
          const float tt  = copysignf(1.0f / (fabsf(tau) + sqrtf(1.0f + tau * tau)), tau);
          c = rsqrtf(1.0f + tt * tt);
          s = tt * c;
        }
        pA[t] = p; qA[t] = q; cA[t] = c; sA[t] = s;
      }
      __syncthreads();
      const int k = t & 63, h = t >> 6;
      const int p = pA[k], q = qA[k];
      const float c = cA[k], s = sA[k];
      // Phase B: A <- J^T A  (rows p,q; pairs act on disjoint rows)
      for (int d = 0; d < 64; ++d) {
        const int j = h * 64 + d;
        const float ap = As[p * 129 + j], aq = As[q * 129 + j];
        As[p * 129 + j] = c * ap - s * aq;
        As[q * 129 + j] = s * ap + c * aq;
      }
      __syncthreads();
      // Phase C: A <- A J  and  V <- V J  (columns p,q)
      for (int d = 0; d < 64; ++d) {
        const int i = h * 64 + d;
        const float ap = As[i * 129 + p], aq = As[i * 129 + q];
        As[i * 129 + p] = c * ap - s * aq;
        As[i * 129 + q] = s * ap + c * aq;
        const float vp = Vs[i * 129 + p], vq = Vs[i * 129 + q];
        Vs[i * 129 + p] = c * vp - s * vq;
        Vs[i * 129 + q] = s * vp + c * vq;
      }
      __syncthreads();
    }
  }

  const float e = As[t * 129 + t];
  ev[t] = e;
  __syncthreads();
  int rank = 0;
  for (int u = 0; u < 128; ++u) {
    const float eu = ev[u];
    rank += (eu < e || (eu == e && u < t)) ? 1 : 0;
  }
  evg[b * 128 + rank] = e;
  for (int i = 0; i < 128; ++i)
    Vg[((size_t)b * 128 + i) * 128 + rank] = Vs[i * 129 + t];
}

// ---------------------------------------------------------------------------
// Kernel 4: node projection  nf[11] @ W_out_nodes + b, masked.
// ---------------------------------------------------------------------------
__global__ __launch_bounds__(128) void nodes_out_kernel(
    const float* __restrict__ nodes, const float* __restrict__ wsNE,
    const float* __restrict__ wsEv, const float* __restrict__ Wn,
    const float* __restrict__ bn, const unsigned char* __restrict__ nm,
    float* __restrict__ out) {
  const int b = blockIdx.x, i = threadIdx.x;
  const float sn = sigmoidf(wsEv[b * 128 + i]);
  const float mask = nm[b * 128 + i] ? 1.0f : 0.0f;
  const float* nd = nodes + ((size_t)b * 128 + i) * 5;
  for (int c = 0; c < 5; ++c) {
    float acc = bn[c];
    for (int k = 0; k < 5; ++k) acc += nd[k] * Wn[k * 5 + c];
    for (int k = 0; k < 5; ++k) acc += wsNE[b * 5 + k] * Wn[(5 + k) * 5 + c];
    acc += sn * Wn[10 * 5 + c];
    out[((size_t)b * 128 + i) * 5 + c] = acc * mask;
  }
}

// ---------------------------------------------------------------------------
// Kernel 5: edge projection with fp32 WMMA (V_WMMA_F32_16X16X4_F32 chain).
// Block = (i, b); 8 waves x 16 rows (j) per block.  Symmetrization is folded
// algebraically: edges/emb are (i,j)-symmetric; the eigenvector feature is
// averaged: 0.5*(sigmoid(V[i,j]) + sigmoid(V[j,i])).  Features k=11..15 are
// zero (conv_features) so only K-chunks 0..2 are multiplied.
// A 16x4 f32 layout: lane (m+16h) holds {A[m][2h], A[m][2h+1]}; B mirrored.
// D 16x16 f32: lane (n+16h), VGPR r -> D[r+8h][n].
// ---------------------------------------------------------------------------
__global__ __launch_bounds__(256) void edges_out_kernel(
    const float* __restrict__ edges, const float* __restrict__ Vg,
    const float* __restrict__ wsEE, const float* __restrict__ We,
    const float* __restrict__ be, const unsigned char* __restrict__ nm,
    float* __restrict__ out) {
  const int i = blockIdx.x, b = blockIdx.y;
  const int tid = threadIdx.x;
  const int wave = tid >> 5, lane = tid & 31;
  const int m = lane & 15, h = lane >> 4;
  const int j = wave * 16 + m;

  // Feature row F[0..11] for (b,i,j): edges(5) | edges_emb(5) | spec(1) | 0
  const size_t eb = (((size_t)b * 128 + i) * 128 + j) * 5;
  float F[12];
  F[0] = edges[eb + 0]; F[1] = edges[eb + 1]; F[2] = edges[eb + 2];
  F[3] = edges[eb + 3]; F[4] = edges[eb + 4];
  F[5] = wsEE[b * 5 + 0]; F[6] = wsEE[b * 5 + 1]; F[7] = wsEE[b * 5 + 2];
  F[8] = wsEE[b * 5 + 3]; F[9] = wsEE[b * 5 + 4];
  const float vij = Vg[((size_t)b * 128 + i) * 128 + j];
  const float vji = Vg[((size_t)b * 128 + j) * 128 + i];
  F[10] = 0.5f * (sigmoidf(vij) + sigmoidf(vji));
  F[11] = 0.0f;

  v2f a0 = { F[0 + 2 * h], F[1 + 2 * h] };
  v2f a1 = { F[4 + 2 * h], F[5 + 2 * h] };
  v2f a2 = { F[8 + 2 * h], F[9 + 2 * h] };

  auto WVL = [&](int k) -> float { return (m < 5) ? We[k * 5 + m] : 0.0f; };
  v2f b0 = { WVL(0 + 2 * h), WVL(1 + 2 * h) };
  v2f b1 = { WVL(4 + 2 * h), WVL(5 + 2 * h) };
  v2f b2 = { WVL(8 + 2 * h), WVL(9 + 2 * h) };

  v8f d = {0.f, 0.f, 0.f, 0.f, 0.f, 0.f, 0.f, 0.f};
  d = __builtin_amdgcn_wmma_f32_16x16x4_f32(false, a0, false, b0, (short)0, d, false, false);
  d = __builtin_amdgcn_wmma_f32_16x16x4_f32(false, a1, false, b1, (short)0, d, false, false);
  d = __builtin_amdgcn_wmma_f32_16x16x4_f32(false, a2, false, b2, (short)0, d, false, false);

  if (m < 5) {  // lane's column n == m; only 5 output classes
    const float nmi = nm[b * 128 + i] ? 1.0f : 0.0f;
    for (int r = 0; r < 8; ++r) {
      const int jr = wave * 16 + r + 8 * h;
      const float nmj = nm[b * 128 + jr] ? 1.0f : 0.0f;
      const float val = (d[r] + be[m]) * nmi * nmj;
      out[(((size_t)b * 128 + i) * 128 + jr) * 5 + m] = val;
    }
  }
}

// ---------------------------------------------------------------------------
extern "C" void kernel_launch(void* const* d_in, const int* in_sizes, int n_in,
                              void* d_out, int out_size, void* d_ws, size_t ws_size,
                              hipStream_t stream) {
  (void)in_sizes; (void)n_in; (void)out_size; (void)ws_size;
  const float* nodes        = (const float*)d_in[0];
  const float* edges        = (const float*)d_in[1];
  const unsigned char* nmsk = (const unsigned char*)d_in[2];
  const float* embedding    = (const float*)d_in[3];
  const float* W_emb_nodes  = (const float*)d_in[4];
  const float* b_emb_nodes  = (const float*)d_in[5];
  const float* W_emb_edges  = (const float*)d_in[6];
  const float* b_emb_edges  = (const float*)d_in[7];
  const float* W_out_nodes  = (const float*)d_in[8];
  const float* b_out_nodes  = (const float*)d_in[9];
  const float* W_out_edges  = (const float*)d_in[10];
  const float* b_out_edges  = (const float*)d_in[11];

  float* out_nodes = (float*)d_out;
  float* out_edges = out_nodes + (size_t)256 * 128 * 5;

  float* wsA  = (float*)d_ws;                       // 256*128*128 laplacians
  float* wsV  = wsA  + (size_t)256 * 128 * 128;     // 256*128*128 eigenvectors
  float* wsEv = wsV  + (size_t)256 * 128 * 128;     // 256*128 eigenvalues (sorted)
  float* wsNE = wsEv + 256 * 128;                   // 256*5 node emb (post-sigmoid)
  float* wsEE = wsNE + 256 * 5;                     // 256*5 edge emb (post-sigmoid)

  emb_kernel<<<256, 32, 0, stream>>>(embedding, W_emb_nodes, b_emb_nodes,
                                     W_emb_edges, b_emb_edges, wsNE, wsEE);
  lap_kernel<<<dim3(128, 256), 128, 0, stream>>>(edges, wsA);
  jacobi_kernel<<<256, 128, 0, stream>>>(wsA, wsV, wsEv);
  nodes_out_kernel<<<256, 128, 0, stream>>>(nodes, wsNE, wsEv, W_out_nodes,
                                            b_out_nodes, nmsk, out_nodes);
  edges_out_kernel<<<dim3(128, 256), 256, 0, stream>>>(edges, wsV, wsEE,
                                                       W_out_edges, b_out_edges,
                                                       nmsk, out_edges);
}